// BaichuanAttention_39006892982807
// MI455X (gfx1250) — compile-verified
//
#include <hip/hip_runtime.h>
#include <hip/hip_bf16.h>

typedef __attribute__((ext_vector_type(2))) float v2f;
typedef __attribute__((ext_vector_type(8))) float v8f;

#define HIDDEN   5120
#define NHEADS   40
#define HDIM     128
#define MAXLEN   2048
#define QLEN     8
#define QKV      (3 * HIDDEN)   // 15360
#define SCALE    0.08838834764831845f   // 1/sqrt(128)
#define FMIN     (-3.402823466e38f)

// ---------------------------------------------------------------------------
// Skinny GEMM: part[s][8][N] = A[8][K] @ W[N][K]^T over K-slice s.
// One wave handles TWO 16-column tiles (32 W rows, lane -> row j0+lane).
// Rows padded to 16; rows 8-15 alias 0-7 and are masked to zero.
// f32 WMMA 16x16x4 layouts:
//   A: lanes 0-15 -> K=k,k+1 ; lanes 16-31 -> K=k+2,k+3 (row = lane&15)
//   B: lane n holds column n; lo lanes K=k,k+1, hi lanes K=k+2,k+3
// K unrolled by 32: 24 b128 loads batched ahead of 16 WMMAs (2 indep chains).
// ---------------------------------------------------------------------------
__global__ __launch_bounds__(32) void gemm8_wmma_kernel(
    const float* __restrict__ A, const float* __restrict__ W,
    float* __restrict__ part, int K, int N, int KS) {
  const int j0   = blockIdx.x * 32;
  const int s    = blockIdx.y;
  const int lane = (int)threadIdx.x;
  const int ln   = lane & 15;
  const int hi   = lane >> 4;
  const float* arow  = A + (size_t)(ln & 7) * K;     // rows 8-15 alias 0-7
  const float  am    = (ln < 8) ? 1.0f : 0.0f;       // ...and are zeroed
  const float* wrow0 = W + (size_t)(j0 + ln) * K;
  const float* wrow1 = W + (size_t)(j0 + 16 + ln) * K;
  const float* wlast = W + (size_t)N * K - 4;        // clamp for prefetch
  const int k0 = s * KS, k1 = k0 + KS;
  v8f acc0 = {}, acc1 = {};
  for (int k = k0; k < k1; k += 32) {
    float4 av[8], wv0[8], wv1[8];
#pragma unroll
    for (int u = 0; u < 8; ++u) {
      av[u]  = *(const float4*)(arow  + k + 4 * u);
      wv0[u] = *(const float4*)(wrow0 + k + 4 * u);
      wv1[u] = *(const float4*)(wrow1 + k + 4 * u);
    }
    // stream the big weight rows ahead (global_prefetch_b8); clamp in-bounds
    const float* pf0 = wrow0 + k + 256;
    const float* pf1 = wrow1 + k + 256;
    __builtin_prefetch((pf0 <= wlast) ? pf0 : wlast, 0, 3);
    __builtin_prefetch((pf1 <= wlast) ? pf1 : wlast, 0, 3);
#pragma unroll
    for (int u = 0; u < 8; ++u) {
      v2f a, b0, b1;
      a.x  = am * (hi ? av[u].z : av[u].x);
      a.y  = am * (hi ? av[u].w : av[u].y);
      b0.x = hi ? wv0[u].z : wv0[u].x;
      b0.y = hi ? wv0[u].w : wv0[u].y;
      b1.x = hi ? wv1[u].z : wv1[u].x;
      b1.y = hi ? wv1[u].w : wv1[u].y;
      acc0 = __builtin_amdgcn_wmma_f32_16x16x4_f32(false, a, false, b0,
                                                   (short)0, acc0, false, false);
      acc1 = __builtin_amdgcn_wmma_f32_16x16x4_f32(false, a, false, b1,
                                                   (short)0, acc1, false, false);
    }
  }
  if (lane < 16) {                       // D rows 0-7 live in lanes 0-15
    float* po = part + (size_t)s * 8 * N;
#pragma unroll
    for (int r = 0; r < 8; ++r) {
      po[(size_t)r * N + j0 + lane]      = acc0[r];
      po[(size_t)r * N + j0 + 16 + lane] = acc1[r];
    }
  }
}

// out[i] = sum of 4 K-slice partials (deterministic reduction, no atomics)
__global__ __launch_bounds__(256) void reduce4_kernel(
    const float* __restrict__ part, float* __restrict__ out, int n) {
  const int i = blockIdx.x * 256 + (int)threadIdx.x;
  if (i < n)
    out[i] = part[i] + part[(size_t)n + i] + part[2 * (size_t)n + i] +
             part[3 * (size_t)n + i];
}

// ---------------------------------------------------------------------------
// scores[h][q][p] = (Q[h,q,:] . Kcache[h,p,:]) * SCALE + mask[h, ipos[q], p]
// One wave per (head, 16-position tile).  K = HDIM = 128, unrolled by 32.
// ---------------------------------------------------------------------------
__global__ __launch_bounds__(32) void scores_wmma_kernel(
    const float* __restrict__ proj, const float* __restrict__ kcache,
    const float* __restrict__ mask, const int* __restrict__ ipos,
    float* __restrict__ S) {
  const int h    = blockIdx.x;
  const int p0   = blockIdx.y * 16;
  const int lane = (int)threadIdx.x;
  const int ln   = lane & 15;
  const int hi   = lane >> 4;
  const float* qrow = proj + (size_t)(ln & 7) * QKV + h * HDIM;
  const float  qm   = (ln < 8) ? 1.0f : 0.0f;
  const float* krow = kcache + ((size_t)h * MAXLEN + (p0 + ln)) * HDIM;
  v8f acc = {};
  for (int k = 0; k < HDIM; k += 32) {
    float4 qv[8], kv[8];
#pragma unroll
    for (int u = 0; u < 8; ++u) {
      qv[u] = *(const float4*)(qrow + k + 4 * u);
      kv[u] = *(const float4*)(krow + k + 4 * u);
    }
#pragma unroll
    for (int u = 0; u < 8; ++u) {
      v2f a, b;
      a.x = qm * (hi ? qv[u].z : qv[u].x);
      a.y = qm * (hi ? qv[u].w : qv[u].y);
      b.x = hi ? kv[u].z : kv[u].x;
      b.y = hi ? kv[u].w : kv[u].y;
      acc = __builtin_amdgcn_wmma_f32_16x16x4_f32(false, a, false, b,
                                                  (short)0, acc, false, false);
    }
  }
  if (lane < 16) {
    const int p = p0 + lane;
#pragma unroll
    for (int r = 0; r < 8; ++r) {
      const int qp = ipos[r];
      float sc = acc[r] * SCALE + mask[((size_t)h * MAXLEN + qp) * MAXLEN + p];
      S[((size_t)h * QLEN + r) * MAXLEN + p] = fmaxf(sc, FMIN);
    }
  }
}

// Overwrite the 8 cached-K columns that the decode step replaces:
// S[h,q,ipos[i]] = (Q[h,q,:] . Knew[h,i,:]) * SCALE + mask[h, ipos[q], ipos[i]]
__global__ __launch_bounds__(64) void scores_fix_kernel(
    const float* __restrict__ proj, const float* __restrict__ mask,
    const int* __restrict__ ipos, float* __restrict__ S) {
  const int h = blockIdx.x;
  const int q = (int)threadIdx.x >> 3;
  const int i = (int)threadIdx.x & 7;
  const float* qrow = proj + (size_t)q * QKV + h * HDIM;
  const float* krow = proj + (size_t)i * QKV + HIDDEN + h * HDIM;
  float acc = 0.f;
#pragma unroll 4
  for (int d = 0; d < HDIM; ++d) acc = fmaf(qrow[d], krow[d], acc);
  const int p = ipos[i];
  float sc = acc * SCALE + mask[((size_t)h * MAXLEN + ipos[q]) * MAXLEN + p];
  S[((size_t)h * QLEN + q) * MAXLEN + p] = fmaxf(sc, FMIN);
}

// Row softmax over 2048, one 256-thread block (8 wave32) per (h,q) row.
__global__ __launch_bounds__(256) void softmax_kernel(float* __restrict__ S) {
  float* s = S + (size_t)blockIdx.x * MAXLEN;
  const int t = (int)threadIdx.x;
  __shared__ float red[8];
  float v[8];
  float mx = FMIN;
#pragma unroll
  for (int e = 0; e < 8; ++e) { v[e] = s[t + e * 256]; mx = fmaxf(mx, v[e]); }
  for (int off = 16; off; off >>= 1) mx = fmaxf(mx, __shfl_xor(mx, off, 32));
  if ((t & 31) == 0) red[t >> 5] = mx;
  __syncthreads();
  mx = red[0];
#pragma unroll
  for (int w = 1; w < 8; ++w) mx = fmaxf(mx, red[w]);
  __syncthreads();
  float sum = 0.f;
#pragma unroll
  for (int e = 0; e < 8; ++e) { v[e] = __expf(v[e] - mx); sum += v[e]; }
  for (int off = 16; off; off >>= 1) sum += __shfl_xor(sum, off, 32);
  if ((t & 31) == 0) red[t >> 5] = sum;
  __syncthreads();
  sum = 0.f;
#pragma unroll
  for (int w = 0; w < 8; ++w) sum += red[w];
  const float inv = 1.0f / sum;
#pragma unroll
  for (int e = 0; e < 8; ++e) s[t + e * 256] = v[e] * inv;
}

// ---------------------------------------------------------------------------
// O_part[z][q][h*128+d] = P[h,q,kslice] @ Vcache[h,kslice,d-tile]
// B = V directly (no transpose): lane n holds column d0+n.  K unrolled by 32.
// ---------------------------------------------------------------------------
__global__ __launch_bounds__(32) void attn_wmma_kernel(
    const float* __restrict__ P, const float* __restrict__ vcache,
    float* __restrict__ part) {
  const int h    = blockIdx.x;
  const int d0   = blockIdx.y * 16;
  const int z    = blockIdx.z;                 // 512-position K slice
  const int lane = (int)threadIdx.x;
  const int ln   = lane & 15;
  const int hi   = lane >> 4;
  const float* prow = P + ((size_t)h * QLEN + (ln & 7)) * MAXLEN;
  const float  pm   = (ln < 8) ? 1.0f : 0.0f;
  const float* vb   = vcache + (size_t)h * MAXLEN * HDIM + d0 + ln;
  const int k0 = z * 512, k1 = k0 + 512;
  v8f acc = {};
  for (int k = k0; k < k1; k += 32) {
    float2 pv[8];
    float  bx[8], by[8];
#pragma unroll
    for (int u = 0; u < 8; ++u) {
      const int kk = k + 4 * u + 2 * hi;
      pv[u] = *(const float2*)(prow + kk);
      bx[u] = vb[(size_t)kk * HDIM];
      by[u] = vb[(size_t)(kk + 1) * HDIM];
    }
#pragma unroll
    for (int u = 0; u < 8; ++u) {
      v2f a, b;
      a.x = pm * pv[u].x;
      a.y = pm * pv[u].y;
      b.x = bx[u];
      b.y = by[u];
      acc = __builtin_amdgcn_wmma_f32_16x16x4_f32(false, a, false, b,
                                                  (short)0, acc, false, false);
    }
  }
  if (lane < 16) {
    float* po = part + (size_t)z * QLEN * HIDDEN;
#pragma unroll
    for (int r = 0; r < 8; ++r)
      po[(size_t)r * HIDDEN + h * HDIM + d0 + lane] = acc[r];
  }
}

// V correction for the 8 replaced cache rows:
// O[q, h*128+d] += sum_i P[h,q,ipos[i]] * (Vnew[h,i,d] - Vcache[h,ipos[i],d])
__global__ __launch_bounds__(256) void attn_fix_kernel(
    const float* __restrict__ P, const float* __restrict__ proj,
    const float* __restrict__ vcache, const int* __restrict__ ipos,
    float* __restrict__ O) {
  const int h = blockIdx.x;
  const int t = (int)threadIdx.x;
  __shared__ int   ip[8];
  __shared__ float pw[64];
  if (t < 8) ip[t] = ipos[t];
  __syncthreads();
  if (t < 64) pw[t] = P[((size_t)h * QLEN + (t >> 3)) * MAXLEN + ip[t & 7]];
  __syncthreads();
#pragma unroll
  for (int e = 0; e < 4; ++e) {
    const int idx = t + e * 256;
    const int q = idx >> 7, d = idx & 127;
    float corr = 0.f;
#pragma unroll
    for (int i = 0; i < 8; ++i) {
      const float vold = vcache[((size_t)h * MAXLEN + ip[i]) * HDIM + d];
      const float vnew = proj[(size_t)i * QKV + 2 * HIDDEN + h * HDIM + d];
      corr = fmaf(pw[q * 8 + i], vnew - vold, corr);
    }
    O[(size_t)q * HIDDEN + h * HDIM + d] += corr;
  }
}

extern "C" void kernel_launch(void* const* d_in, const int* in_sizes, int n_in,
                              void* d_out, int out_size, void* d_ws,
                              size_t ws_size, hipStream_t stream) {
  (void)in_sizes; (void)n_in; (void)out_size; (void)ws_size;
  const int*   ipos   = (const int*)  d_in[0];
  const float* hidden = (const float*)d_in[1];
  const float* mask   = (const float*)d_in[2];
  const float* Wp     = (const float*)d_in[3];
  const float* Wo     = (const float*)d_in[4];
  const float* kcache = (const float*)d_in[5];
  const float* vcache = (const float*)d_in[6];
  float* out = (float*)d_out;

  // workspace layout (floats): total 1,310,720 f32 = 5.0 MB
  float* ws   = (float*)d_ws;
  float* proj = ws;                         // [8][15360]          122880
  float* S    = proj + QLEN * QKV;          // [40][8][2048]       655360
  float* O    = S + NHEADS * QLEN * MAXLEN; // [8][5120]            40960
  float* part = O + QLEN * HIDDEN;          // scratch partials    491520

  // 1) fused QKV projection (2 tiles/wave, K split 4-way)
  gemm8_wmma_kernel<<<dim3(QKV / 32, 4), 32, 0, stream>>>(
      hidden, Wp, part, HIDDEN, QKV, HIDDEN / 4);
  reduce4_kernel<<<dim3((QLEN * QKV + 255) / 256), 256, 0, stream>>>(
      part, proj, QLEN * QKV);

  // 2) scores = Q K^T * scale + mask (all 2048 positions), then patch the
  //    8 decode positions with the freshly-projected K (cache not mutable)
  scores_wmma_kernel<<<dim3(NHEADS, MAXLEN / 16), 32, 0, stream>>>(
      proj, kcache, mask, ipos, S);
  scores_fix_kernel<<<dim3(NHEADS), 64, 0, stream>>>(proj, mask, ipos, S);

  // 3) row softmax
  softmax_kernel<<<dim3(NHEADS * QLEN), 256, 0, stream>>>(S);

  // 4) O = P @ V over original cache (K split 4-way), then exact correction
  //    for the 8 replaced V rows
  attn_wmma_kernel<<<dim3(NHEADS, HDIM / 16, 4), 32, 0, stream>>>(
      S, vcache, part);
  reduce4_kernel<<<dim3((QLEN * HIDDEN + 255) / 256), 256, 0, stream>>>(
      part, O, QLEN * HIDDEN);
  attn_fix_kernel<<<dim3(NHEADS), 256, 0, stream>>>(S, proj, vcache, ipos, O);

  // 5) output projection (2 tiles/wave, K split 4-way)
  gemm8_wmma_kernel<<<dim3(HIDDEN / 32, 4), 32, 0, stream>>>(
      O, Wo, part, HIDDEN, HIDDEN, HIDDEN / 4);
  reduce4_kernel<<<dim3((QLEN * HIDDEN + 255) / 256), 256, 0, stream>>>(
      part, out, QLEN * HIDDEN);
}